// FC_Classifier_38439957299602
// MI455X (gfx1250) — compile-verified
//
#include <hip/hip_runtime.h>
#include <hip/hip_bf16.h>

// ---------------------------------------------------------------------------
// Problem constants (from reference setup_inputs)
// ---------------------------------------------------------------------------
#define C_DIM   2048            // channels (GEMM K)
#define H_DIM   192
#define W_DIM   192
#define PIX     (H_DIM * W_DIM) // 36864 pixels (GEMM N)
#define NC      150             // classes (GEMM M)
#define NCP     160             // classes padded to 10 WMMA tiles of 16
#define JT      (NCP / 16)      // 10 class tiles per wave
#define TS      152             // j-stride (floats) per (y,x) cell of integral T
#define TCELLS  (193 * 193)     // integral image cells per channel

// GEMM K-chunking for LDS-staged A
#define KC      32              // K values per LDS chunk
#define KLP     36              // padded LDS row stride (floats): bank-conflict
                                // free (gcd(36,64)=4, 16-lane cycle) and 16B
                                // aligned rows (36*4 = 144 = 9*16)
#define NCHUNK  (C_DIM / KC)    // 64 chunks

typedef __attribute__((ext_vector_type(2))) float v2f;
typedef __attribute__((ext_vector_type(8))) float v8f;

// ---------------------------------------------------------------------------
// 1) Zero-pad fc_w (150x2048) -> A_pad (160x2048) so the GEMM never branches.
// ---------------------------------------------------------------------------
__global__ void padw_kernel(const float* __restrict__ fcw, float* __restrict__ Ap) {
    size_t i = (size_t)blockIdx.x * blockDim.x + threadIdx.x;  // < 160*2048
    int j = (int)(i / C_DIM);
    int c = (int)(i % C_DIM);
    Ap[i] = (j < NC) ? fcw[(size_t)j * C_DIM + c] : 0.0f;
}

// ---------------------------------------------------------------------------
// 2) Box index computation (replicates reference _box_indices exactly).
// ---------------------------------------------------------------------------
__global__ void boxidx_kernel(const float* __restrict__ anchors,
                              const float* __restrict__ scale,
                              int* __restrict__ x0o, int* __restrict__ x1o,
                              int* __restrict__ y0o, int* __restrict__ y1o,
                              float* __restrict__ inva, int N) {
    int n = blockIdx.x * blockDim.x + threadIdx.x;
    if (n >= N) return;
    float s0 = scale[0], s1 = scale[1];
    float a0 = anchors[n * 4 + 0] * 0.125f;  // /DS, DS=8
    float a1 = anchors[n * 4 + 1] * 0.125f;
    float a2 = anchors[n * 4 + 2] * 0.125f;
    float a3 = anchors[n * 4 + 3] * 0.125f;
    int x0 = (int)(a0 * s1);  // C truncation == astype(int32) for non-neg
    int x1 = (int)(a1 * s1);
    int y0 = (int)(a2 * s0);
    int y1 = (int)(a3 * s0);
    bool eqy = (y0 == y1);
    int y0n = (eqy && y0 != 0) ? y0 - 1 : y0;
    int y1n = (eqy && y0 == 0) ? y1 + 1 : y1;
    bool eqx = (x0 == x1);
    int x0n = (eqx && x0 != 0) ? x0 - 1 : x0;
    int x1n = (eqx && x0 == 0) ? x1 + 1 : x1;
    y0n = min(max(y0n, 0), H_DIM); y1n = min(max(y1n, 0), H_DIM);
    x0n = min(max(x0n, 0), W_DIM); x1n = min(max(x1n, 0), W_DIM);
    int area = (y1n - y0n) * (x1n - x0n);
    if (area < 1) area = 1;
    x0o[n] = x0n; x1o[n] = x1n; y0o[n] = y0n; y1o[n] = y1n;
    inva[n] = 1.0f / (float)area;
}

// ---------------------------------------------------------------------------
// 3) f32 WMMA GEMM:  G[p][j] = sum_c A_pad[j][c] * fm[c][p]
//    - One wave per 16-pixel column; all 10 class tiles in registers.
//    - A staged in LDS per block via async global->LDS b128 copies
//      (ASYNCcnt), double-buffered so the copy of chunk c+1 overlaps the
//      80 WMMAs of chunk c.  Frags then come from DS loads, keeping the
//      VMEM pipe free to stream the 302 MB feature map from HBM.
//    - All 10 A-fragments of a K-step are hoisted into registers before the
//      WMMA chain so multiple DS loads stay in flight (partial dscnt waits)
//      instead of one load->wait->wmma round trip per tile.
//    V_WMMA_F32_16X16X4_F32 layouts (wave32):
//      A (16x4):  lane<16 -> {K=0,K=1}, lane>=16 -> {K=2,K=3}, M = lane&15
//      B (4x16):  lane<16 -> rows K=0/1, lane>=16 -> rows K=2/3, N = lane&15
//      C/D:       VGPR r: M = r + 8*(lane>>4), N = lane&15
// ---------------------------------------------------------------------------
__global__ __launch_bounds__(256) void gemm_wmma_kernel(
        const float* __restrict__ fm,   // [C_DIM][PIX]
        const float* __restrict__ Ap,   // [NCP][C_DIM]
        float* __restrict__ G) {        // [PIX][NCP]
    __shared__ float ldsA[2][NCP * KLP];          // 2 x 22.5 KB

    const int tid   = threadIdx.x;
    const int lane  = tid & 31;
    const int wid   = tid >> 5;
    const int pbase = (blockIdx.x * 8 + wid) * 16;   // pixel tile base
    const int n     = lane & 15;                     // N index / M row of A
    const int hi    = lane >> 4;                     // K half selector

    const v8f vzero = {0.f, 0.f, 0.f, 0.f, 0.f, 0.f, 0.f, 0.f};
    v8f acc[JT];
#pragma unroll
    for (int t = 0; t < JT; ++t) acc[t] = vzero;

    // Per-thread async copy plan: 160x32 floats = 1280 b128 units / 256 thr
    // = 5 units each.  Unit r: row j = (tid>>3) + 32*r, col c4 = (tid&7)*4.
    const int jrow = tid >> 3;
    const int c4   = (tid & 7) << 2;
    const float* gcopy = Ap + (size_t)jrow * C_DIM + c4;
    const unsigned ldsbase0 =
        (unsigned)(size_t)(&ldsA[0][0]) + (unsigned)jrow * (KLP * 4) + (unsigned)c4 * 4;
    const unsigned ldsbase1 =
        (unsigned)(size_t)(&ldsA[1][0]) + (unsigned)jrow * (KLP * 4) + (unsigned)c4 * 4;

    auto issue_copy = [&](int buf, int kb) {
        unsigned lb = buf ? ldsbase1 : ldsbase0;
        const float* g = gcopy + kb;
#pragma unroll
        for (int r = 0; r < 5; ++r) {
            unsigned long long ga =
                (unsigned long long)(size_t)(g + (size_t)r * 32 * C_DIM);
            unsigned la = lb + (unsigned)r * (32u * KLP * 4u);
            asm volatile("global_load_async_to_lds_b128 %0, %1, off"
                         :: "v"(la), "v"(ga) : "memory");
        }
    };

    issue_copy(0, 0);

    for (int c = 0; c < NCHUNK; ++c) {
        if (c + 1 < NCHUNK) {
            issue_copy((c + 1) & 1, (c + 1) * KC);
            // Async loads complete in order: <=5 outstanding => chunk c landed.
            asm volatile("s_wait_asynccnt 5" ::: "memory");
        } else {
            asm volatile("s_wait_asynccnt 0" ::: "memory");
        }
        __syncthreads();

        const float* lA   = &ldsA[c & 1][0] + 2 * hi;
        const float* bcol = fm + (size_t)c * KC * PIX + pbase + n;
#pragma unroll
        for (int kk = 0; kk < KC; kk += 4) {
            const float* bp = bcol + (size_t)(kk + 2 * hi) * PIX;
            v2f b;
            b.x = bp[0];
            b.y = bp[PIX];
            // Hoist all 10 fragments: lets the compiler clause the DS loads
            // and overlap their latency with the WMMA chain below.
            v2f a[JT];
#pragma unroll
            for (int t = 0; t < JT; ++t) {
                a[t] = *(const v2f*)(lA + (t * 16 + n) * KLP + kk);
            }
#pragma unroll
            for (int t = 0; t < JT; ++t) {
                acc[t] = __builtin_amdgcn_wmma_f32_16x16x4_f32(
                    false, a[t], false, b, (short)0, acc[t], false, false);
            }
        }
        __syncthreads();   // safe to overwrite this buffer next iteration
    }

    // Store D: G[(pbase+n)][jt*16 + r + 8*hi]  (j-innermost layout -> the
    // 4-float runs merge into global_store_b128 clauses).
    float* gp = G + (size_t)(pbase + n) * NCP;
#pragma unroll
    for (int t = 0; t < JT; ++t) {
#pragma unroll
        for (int r = 0; r < 8; ++r) {
            gp[t * 16 + r + 8 * hi] = acc[t][r];
        }
    }
}

// ---------------------------------------------------------------------------
// 4) Integral image, pass 1: per-row prefix sums along x.
//    T layout: T[(y*193 + x)*TS + j], y,x in [0,192].  Rows 1..192 filled here;
//    column x=0 zeroed.  Lanes map to j -> fully coalesced.
// ---------------------------------------------------------------------------
__global__ void rowscan_kernel(const float* __restrict__ G, float* __restrict__ T) {
    const int j  = threadIdx.x;       // 0..151
    const int yy = blockIdx.x + 1;    // 1..192
    const float* g = G + (size_t)(yy - 1) * W_DIM * NCP + j;
    float* t = T + (size_t)yy * 193 * TS + j;
    float acc = 0.0f;
    t[0] = 0.0f;                      // x = 0 column
    for (int x = 1; x <= W_DIM; ++x) {
        acc += g[(size_t)(x - 1) * NCP];
        t[(size_t)x * TS] = acc;
    }
}

// ---------------------------------------------------------------------------
// 5) Integral image, pass 2: in-place prefix sums down y.  Row y=0 zeroed.
// ---------------------------------------------------------------------------
__global__ void colscan_kernel(float* __restrict__ T) {
    const int j = threadIdx.x;   // 0..151
    const int x = blockIdx.x;    // 0..192
    float* t = T + (size_t)x * TS + j;
    t[0] = 0.0f;                 // y = 0 row
    float acc = 0.0f;
    for (int y = 1; y <= H_DIM; ++y) {
        size_t o = (size_t)y * 193 * TS;
        acc += t[o];
        t[o] = acc;
    }
}

// ---------------------------------------------------------------------------
// 6) Gather 4 corners + scale by 1/area + bias.  Block = one box, lanes = j.
//    All four corner reads are 150 contiguous floats -> coalesced, L2-hot.
// ---------------------------------------------------------------------------
__global__ void gather_kernel(const float* __restrict__ T,
                              const int* __restrict__ x0a, const int* __restrict__ x1a,
                              const int* __restrict__ y0a, const int* __restrict__ y1a,
                              const float* __restrict__ inva,
                              const float* __restrict__ fcb,
                              float* __restrict__ out) {
    const int n = blockIdx.x;
    const int j = threadIdx.x;
    if (j >= NC) return;
    const int x0 = x0a[n], x1 = x1a[n], y0 = y0a[n], y1 = y1a[n];
    const float* t = T + j;
    float s = t[((size_t)y1 * 193 + x1) * TS]
            - t[((size_t)y0 * 193 + x1) * TS]
            - t[((size_t)y1 * 193 + x0) * TS]
            + t[((size_t)y0 * 193 + x0) * TS];
    out[(size_t)n * NC + j] = s * inva[n] + fcb[j];
}

// ---------------------------------------------------------------------------
// Launch
// ---------------------------------------------------------------------------
extern "C" void kernel_launch(void* const* d_in, const int* in_sizes, int n_in,
                              void* d_out, int out_size, void* d_ws, size_t ws_size,
                              hipStream_t stream) {
    const float* fm      = (const float*)d_in[0];  // (2048,192,192) f32
    const float* scale   = (const float*)d_in[1];  // (2,) f32
    const float* anchors = (const float*)d_in[2];  // (N,4) f32
    const float* fcw     = (const float*)d_in[3];  // (150,2048) f32
    const float* fcb     = (const float*)d_in[4];  // (150,) f32
    const int N = in_sizes[2] / 4;                 // 16384

    // Workspace carving (floats).
    float* ws = (float*)d_ws;
    size_t off = 0;
    float* Ap = ws + off;  off += (size_t)NCP * C_DIM;          // 327,680
    int*   x0 = (int*)(ws + off); off += N;
    int*   x1 = (int*)(ws + off); off += N;
    int*   y0 = (int*)(ws + off); off += N;
    int*   y1 = (int*)(ws + off); off += N;
    float* inva = ws + off; off += N;
    off = (off + 63) & ~(size_t)63;                             // align
    float* G = ws + off;   off += (size_t)PIX * NCP;            // 23.6 MB
    float* T = ws + off;   off += (size_t)TCELLS * TS;          // 22.6 MB
    (void)ws_size; (void)n_in; (void)out_size;

    padw_kernel<<<(NCP * C_DIM) / 256, 256, 0, stream>>>(fcw, Ap);
    boxidx_kernel<<<(N + 255) / 256, 256, 0, stream>>>(anchors, scale,
                                                       x0, x1, y0, y1, inva, N);
    gemm_wmma_kernel<<<PIX / (16 * 8), 256, 0, stream>>>(fm, Ap, G);
    rowscan_kernel<<<H_DIM, TS, 0, stream>>>(G, T);
    colscan_kernel<<<193, TS, 0, stream>>>(T);
    gather_kernel<<<N, NCP, 0, stream>>>(T, x0, x1, y0, y1, inva, fcb,
                                         (float*)d_out);
}